// DySample_67448166416689
// MI455X (gfx1250) — compile-verified
//
#include <hip/hip_runtime.h>

typedef __attribute__((ext_vector_type(2))) float v2f;
typedef __attribute__((ext_vector_type(8))) float v8f;

#define B_   16
#define C_   256
#define H_   64
#define W_   64
#define HW_  (H_ * W_)      // 4096
#define HR_  128
#define WR_  128
#define HRWR_ (HR_ * WR_)   // 16384

// ---------------------------------------------------------------------------
// Kernel 1: 1x1 conv as a skinny GEMM on the fp32 WMMA pipe.
//   off[b][o][s] = sum_c x[b][c][s] * wgt[o][c] + bias[o],  o in [0,8)
// One wave computes a 16(M=spatial) x 16(N=outputs, 8 valid) tile, K=256 in
// 64 steps of V_WMMA_F32_16X16X4_F32.
// A layout (32-bit 16x4): lanes 0-15 M=0..15 K={0,1}; lanes 16-31 K={2,3}.
// B layout mirrors (rows K striped over lane halves, N over lanes).
// D layout: VGPR v -> M = v + 8*(lane>=16), N = lane&15.
// Padding of N 8->16 is done branchlessly: clamp the weight row index (always
// a valid address) and scale by a loop-invariant 0/1 mask, so the inner loop
// has no EXEC manipulation around the loads.
// ---------------------------------------------------------------------------
__global__ __launch_bounds__(256) void dysample_offset_wmma(
    const float* __restrict__ x, const float* __restrict__ wgt,
    const float* __restrict__ bias, float* __restrict__ off)
{
  const int lane      = threadIdx.x & 31;
  const int waveInBlk = threadIdx.x >> 5;
  const int tile      = blockIdx.x * 8 + waveInBlk;   // 4096 tiles total
  const int mBase     = tile << 4;                    // 16 spatial rows / tile
  const int b         = mBase >> 12;                  // HW_ = 4096 rows per batch
  const int sBase     = mBase & (HW_ - 1);

  const int mLoc = lane & 15;
  const int kOff = (lane >> 4) << 1;                  // 0 or 2
  const int n    = lane & 15;

  const float* xb = x + (size_t)b * C_ * HW_ + sBase + mLoc;

  // Branchless N-padding: lanes with n>=8 read (valid) row 7 and zero it out.
  const float msk = (n < 8) ? 1.0f : 0.0f;
  const int   nr  = (n < 8) ? n : 7;
  const float* wp = wgt + nr * C_ + kOff;             // 8-byte aligned

  v8f acc = {};
  for (int k0 = 0; k0 < C_; k0 += 4) {
    v2f a, bf;
    a.x = xb[(size_t)(k0 + kOff)     * HW_];
    a.y = xb[(size_t)(k0 + kOff + 1) * HW_];
    bf  = *(const v2f*)(wp + k0);                     // one global_load_b64
    bf.x *= msk;
    bf.y *= msk;
    // 8 args: (neg_a, A, neg_b, B, c_mod, C, reuse_a, reuse_b)
    acc = __builtin_amdgcn_wmma_f32_16x16x4_f32(false, a, false, bf,
                                                (short)0, acc, false, false);
  }

  if (n < 8) {
    const float bo = bias[n];
    float* op = off + ((size_t)b * 8 + n) * HW_ + sBase + ((lane >> 4) << 3);
#pragma unroll
    for (int v = 0; v < 8; ++v) op[v] = acc[v] + bo;
  }
}

// ---------------------------------------------------------------------------
// Kernel 2: pixel-shuffle + grid build (with the reference's h/w axis swap)
// + bilinear zero-padded sampling over all 256 channels.
// Reference swap makes ix ~ f(ho), iy ~ f(wo): gathers coalesce along ho,
// stores along wo -> transpose each 32x32 tile through padded LDS.
// ---------------------------------------------------------------------------
__global__ __launch_bounds__(1024) void dysample_sample(
    const float* __restrict__ x, const float* __restrict__ off,
    float* __restrict__ out)
{
  __shared__ float lds[32 * 33];

  const int tid = threadIdx.x;
  const int b   = blockIdx.x >> 4;          // 16 tiles per batch image
  const int t   = blockIdx.x & 15;
  const int tho = (t >> 2) << 5;            // tile origin in ho
  const int two = (t & 3) << 5;             // tile origin in wo

  // ---- load mapping: fast along ho (coalesced gathers) ----
  const int lho = tid & 31;
  const int lwo = tid >> 5;
  const int ho  = tho + lho;
  const int wo  = two + lwo;

  // pixel-shuffle inverse: (ho,wo) -> conv-output channel + input pixel
  const int h = ho >> 1, ii = ho & 1;
  const int w = wo >> 1, jj = wo & 1;
  const int ch = ii * 2 + jj;
  const int s  = h * W_ + w;
  const float* offb = off + (size_t)b * 8 * HW_;
  const float off0 = offb[ch * HW_ + s];        // -> grid[...,0] (x-coord)
  const float off1 = offb[(4 + ch) * HW_ + s];  // -> grid[...,1] (y-coord)

  const float gh = -1.0f + (2.0f / 127.0f) * (float)ho;
  const float gw = -1.0f + (2.0f / 127.0f) * (float)wo;
  const float gx = gh + off0;  // reference swap: x-coord built from h-range
  const float gy = gw + off1;
  const float ix = (gx + 1.0f) * 32.0f - 0.5f;  // ((gx+1)*W - 1) * 0.5
  const float iy = (gy + 1.0f) * 32.0f - 0.5f;

  const float fx0 = floorf(ix), fy0 = floorf(iy);
  const int   x0  = (int)fx0,   y0  = (int)fy0;
  const float wx1 = ix - fx0,   wy1 = iy - fy0;
  const float wx0 = 1.0f - wx1, wy0 = 1.0f - wy1;
  const int   x1  = x0 + 1,     y1  = y0 + 1;

  const float vx0 = (x0 >= 0 && x0 < W_) ? 1.0f : 0.0f;
  const float vx1 = (x1 >= 0 && x1 < W_) ? 1.0f : 0.0f;
  const float vy0 = (y0 >= 0 && y0 < H_) ? 1.0f : 0.0f;
  const float vy1 = (y1 >= 0 && y1 < H_) ? 1.0f : 0.0f;
  const float w00 = wy0 * wx0 * (vy0 * vx0);
  const float w01 = wy0 * wx1 * (vy0 * vx1);
  const float w10 = wy1 * wx0 * (vy1 * vx0);
  const float w11 = wy1 * wx1 * (vy1 * vx1);

  const int x0c = min(max(x0, 0), W_ - 1);
  const int x1c = min(max(x1, 0), W_ - 1);
  const int y0c = min(max(y0, 0), H_ - 1);
  const int y1c = min(max(y1, 0), H_ - 1);
  const int o00 = y0c * W_ + x0c;
  const int o01 = y0c * W_ + x1c;
  const int o10 = y1c * W_ + x0c;
  const int o11 = y1c * W_ + x1c;

  const float* xb = x + (size_t)b * C_ * HW_;

  // ---- store mapping: fast along wo (coalesced stores) ----
  const int swo = tid & 31;
  const int sho = tid >> 5;
  float* outp = out + (size_t)b * C_ * HRWR_
                    + (size_t)(tho + sho) * WR_ + (two + swo);

  const int wrI = lwo * 33 + lho;  // consecutive lanes -> consecutive banks
  const int rdI = swo * 33 + sho;  // stride 33, gcd(33,64)=1 -> conflict-free

  for (int c = 0; c < C_; ++c) {
    const float* xc = xb + (size_t)c * HW_;
    const float v = w00 * xc[o00] + w01 * xc[o01]
                  + w10 * xc[o10] + w11 * xc[o11];
    lds[wrI] = v;
    __syncthreads();
    outp[(size_t)c * HRWR_] = lds[rdI];
    __syncthreads();
  }
}

// ---------------------------------------------------------------------------
extern "C" void kernel_launch(void* const* d_in, const int* in_sizes, int n_in,
                              void* d_out, int out_size, void* d_ws, size_t ws_size,
                              hipStream_t stream) {
  (void)in_sizes; (void)n_in; (void)out_size; (void)ws_size;
  const float* x    = (const float*)d_in[0];   // (16,256,64,64)
  const float* wgt  = (const float*)d_in[1];   // (8,256)
  const float* bias = (const float*)d_in[2];   // (8,)
  float* out = (float*)d_out;                  // (16,256,128,128)
  float* off = (float*)d_ws;                   // (16,8,64,64) = 2 MiB scratch

  // 4096 M-tiles, 8 waves (256 threads) per block -> 512 blocks
  dysample_offset_wmma<<<512, 256, 0, stream>>>(x, wgt, bias, off);
  // 256 tiles of 32x32 output locations, 1024 threads each
  dysample_sample<<<256, 1024, 0, stream>>>(x, off, out);
}